// SpanModel_53034256171024
// MI455X (gfx1250) — compile-verified
//
#include <hip/hip_runtime.h>

typedef __attribute__((ext_vector_type(2))) float v2f;
typedef __attribute__((ext_vector_type(8))) float v8f;

#define Bv 16
#define Sv 512
#define Hv 768
#define ROWS 32               // span rows per block (2 WMMA row tiles)
#define LDS_STRIDE (Hv + 4)   // 772 floats: rows shift 4 banks -> conflict-free frag reads

// ---------------------------------------------------------------------------
// Kernel A: token_inputs copy + base[b,h] = b_lin[h] + cls[b,:] @ w_lin[H:2H,:]
// ---------------------------------------------------------------------------
__global__ __launch_bounds__(256) void span_cls_base_kernel(
    const float* __restrict__ emb, const float* __restrict__ w_lin,
    const float* __restrict__ b_lin, float* __restrict__ out,
    float* __restrict__ base_ws) {
  int gid = blockIdx.x * 256 + threadIdx.x;           // 48 blocks * 256 = 12288
  int b = gid / Hv;
  int h = gid - b * Hv;
  const float* cls = emb + (size_t)b * Sv * Hv;       // emb[b, 0, :]
  float acc = b_lin[h];
#pragma unroll 8
  for (int f = 0; f < Hv; ++f)
    acc = fmaf(cls[f], w_lin[(size_t)(Hv + f) * Hv + h], acc);
  base_ws[gid] = acc;
  out[gid] = cls[h];                                  // token_inputs output
}

// ---------------------------------------------------------------------------
// Kernel B: span_reps[b,n,h] = (span_max row n) @ w_lin[0:H,:]
//                              + base[b,h] + width(n) * w_lin[2H,h]
// One block = 32 span rows x 768 cols. 8 waves; each wave owns 96 cols
// (6 col tiles) x 2 row tiles; every B fragment feeds 2 WMMAs.
// ---------------------------------------------------------------------------
__global__ __launch_bounds__(256) void span_gemm_kernel(
    const float* __restrict__ emb, const float* __restrict__ w_lin,
    const int* __restrict__ starts, const int* __restrict__ widths,
    const float* __restrict__ base_ws, float* __restrict__ out, int N) {
  __shared__ float lds_a[ROWS * LDS_STRIDE];          // 98.8 KB of 320 KB/WGP

  const int tid = threadIdx.x;
  const int b   = blockIdx.y;
  const int n0  = blockIdx.x * ROWS;
  const float* embB = emb + (size_t)b * Sv * Hv;

  // ---- Phase 1: stage span-max rows. Consecutive spans usually extend the
  // previous span by one token (same start, width+1), so row r is
  // max(row r-1, one new token): ~1 global load per row instead of ~4.5.
  for (int f = tid; f < Hv; f += 256) {               // 3 iterations
    int pst = -1, pw = 0;
    float pm = 0.0f;
#pragma unroll 1
    for (int r = 0; r < ROWS; ++r) {
      int n = n0 + r;
      float m = 0.0f;
      if (n < N) {
        int st = starts[n];
        int w  = widths[n];
        if (st == pst && w == pw + 1) {
          m = fmaxf(pm, embB[(size_t)(st + w - 1) * Hv + f]);
        } else {
          const float* p = embB + (size_t)st * Hv + f;
          m = p[0];
          for (int k = 1; k < w; ++k) m = fmaxf(m, p[(size_t)k * Hv]);
        }
        pst = st; pw = w; pm = m;
      }
      lds_a[r * LDS_STRIDE + f] = m;
    }
  }
  __syncthreads();

  // ---- Phase 2: WMMA GEMM, A from LDS, B from L2-resident w_lin
  const int wave    = tid >> 5;
  const int lane    = tid & 31;
  const int halfsel = lane >> 4;        // 0: K pair {0,1}; 1: K pair {2,3}
  const int lrow    = lane & 15;        // M for A-frag, N for B-frag
  const int koff    = halfsel * 2;
  const int col0    = wave * 96;        // 8 waves * 96 = 768 columns

  v8f acc[2][6];
#pragma unroll
  for (int rt = 0; rt < 2; ++rt)
#pragma unroll
    for (int t = 0; t < 6; ++t) acc[rt][t] = (v8f){};

  const float* arow0 = lds_a + lrow * LDS_STRIDE + koff;        // 8B aligned
  const float* arow1 = arow0 + 16 * LDS_STRIDE;
  for (int k0 = 0; k0 < Hv; k0 += 4) {
    v2f a0 = *(const v2f*)(arow0 + k0);  // A row tile 0, ISA 16x4 layout
    v2f a1 = *(const v2f*)(arow1 + k0);  // A row tile 1
    const float* brow = w_lin + (size_t)(k0 + koff) * Hv + col0 + lrow;
#pragma unroll
    for (int t = 0; t < 6; ++t) {
      v2f bf;
      bf.x = brow[t * 16];               // B[K=k0+koff  ][n]
      bf.y = brow[t * 16 + Hv];          // B[K=k0+koff+1][n]
      acc[0][t] = __builtin_amdgcn_wmma_f32_16x16x4_f32(
          false, a0, false, bf, (short)0, acc[0][t], false, false);
      acc[1][t] = __builtin_amdgcn_wmma_f32_16x16x4_f32(
          false, a1, false, bf, (short)0, acc[1][t], false, false);
    }
  }

  // ---- Epilogue: + base[b,h] + width * w_lin[2H,h]; NT stores (200 MB
  // streamed once; keep w_lin/emb resident in L2). Masked N tail.
  float* outSpan = out + Bv * Hv;                      // after token_inputs
  const size_t rowBase = (size_t)b * (size_t)N * Hv;
  const float* wwid  = w_lin + (size_t)(2 * Hv) * Hv;  // row 1536
  const float* baseB = base_ws + b * Hv;

#pragma unroll
  for (int rt = 0; rt < 2; ++rt) {
    const int nbase = n0 + rt * 16 + halfsel * 8;      // C/D: VGPR j -> M=j+8*halfsel
    float wf[8];
#pragma unroll
    for (int j = 0; j < 8; ++j)
      wf[j] = (nbase + j < N) ? (float)widths[nbase + j] : 0.0f;
#pragma unroll
    for (int t = 0; t < 6; ++t) {
      int nc   = col0 + t * 16 + lrow;
      float cb = baseB[nc];
      float ww = wwid[nc];
#pragma unroll
      for (int j = 0; j < 8; ++j) {
        int n = nbase + j;
        if (n < N) {
          __builtin_nontemporal_store(acc[rt][t][j] + cb + wf[j] * ww,
                                      &outSpan[rowBase + (size_t)n * Hv + nc]);
        }
      }
    }
  }
}

extern "C" void kernel_launch(void* const* d_in, const int* in_sizes, int n_in,
                              void* d_out, int out_size, void* d_ws, size_t ws_size,
                              hipStream_t stream) {
  const float* emb    = (const float*)d_in[0];   // (16, 512, 768) f32
  const float* w_lin  = (const float*)d_in[1];   // (1537, 768) f32
  const float* b_lin  = (const float*)d_in[2];   // (768,) f32
  const int*   starts = (const int*)d_in[3];     // (N,) i32
  const int*   widths = (const int*)d_in[4];     // (N,) i32
  const int N = in_sizes[3];                     // 4068

  float* out     = (float*)d_out;
  float* base_ws = (float*)d_ws;                 // 16*768 f32 = 48 KB scratch

  span_cls_base_kernel<<<(Bv * Hv) / 256, 256, 0, stream>>>(
      emb, w_lin, b_lin, out, base_ws);

  dim3 grid((N + ROWS - 1) / ROWS, Bv);
  span_gemm_kernel<<<grid, 256, 0, stream>>>(
      emb, w_lin, starts, widths, base_ws, out, N);
}